// Memory_27882927686265
// MI455X (gfx1250) — compile-verified
//
#include <hip/hip_runtime.h>

// Sizes fixed by the reference problem.
#define N_CLASS 200
#define N_MU    1024
#define OUT_DIM 512
#define K_IN    512
#define N_TOT   (N_MU + K_IN)    // 1536 candidates per class
#define OUT_ROW (OUT_DIM + 1)    // 513 floats per packed output row
#define ROWS_PER_BLOCK 8         // one wave32 per output row, 8 waves/block

// ---------------------------------------------------------------------------
// Kernel 0: mark which classes are listed in cls_idx (general-case safety;
// with the given inputs this flags all 200 classes).
// ---------------------------------------------------------------------------
__global__ void flags_kernel(const int* __restrict__ cls_idx, int n_idx,
                             int* __restrict__ flag) {
    const int t = threadIdx.x;
    for (int c = t; c < N_CLASS; c += blockDim.x) flag[c] = 0;
    __syncthreads();
    for (int p = t; p < n_idx; p += blockDim.x) {
        const int c = cls_idx[p];
        if (c >= 0 && c < N_CLASS) flag[c] = 1;
    }
}

// ---------------------------------------------------------------------------
// Kernel 1: per-class stable top-1024 of 1536 scores via rank-by-comparison.
// rank(i) = #{j : s[j] > s[i]} + #{j < i : s[j] == s[i]}  (matches
// jax.lax.top_k descending sort with stable ties). Ranks are a permutation,
// so exactly the rank<1024 elements fill all 1024 output slots.
// Writes: perm[c][rank] = candidate index, wsc[c][rank] = score.
// ---------------------------------------------------------------------------
__global__ __launch_bounds__(512) void select_kernel(
    const float* __restrict__ cls_sc_queue,   // [N_CLASS, N_MU]
    const float* __restrict__ inp_sc,         // [K_IN, N_CLASS]
    const int*   __restrict__ flag,           // [N_CLASS]
    int*         __restrict__ perm,           // [N_CLASS, N_MU]
    float*       __restrict__ wsc)            // [N_CLASS, N_MU]
{
    __shared__ float sc[N_TOT];
    const int c = blockIdx.x;
    const int t = threadIdx.x;

    // Stage the 1536 candidate scores in LDS.
    sc[t]        = cls_sc_queue[(size_t)c * N_MU + t];
    sc[t + 512]  = cls_sc_queue[(size_t)c * N_MU + t + 512];
    sc[N_MU + t] = inp_sc[(size_t)t * N_CLASS + c];   // column c of inp_sc
    __syncthreads();

    const size_t base = (size_t)c * N_MU;

    if (flag[c] == 0) {
        // Class not listed: identity permutation, keep old scores.
        perm[base + t]       = t;
        wsc [base + t]       = sc[t];
        perm[base + t + 512] = t + 512;
        wsc [base + t + 512] = sc[t + 512];
        return;
    }

    // Each thread ranks 3 candidates in a single broadcast pass over LDS.
    const int   i0 = t, i1 = t + 512, i2 = t + 1024;
    const float v0 = sc[i0], v1 = sc[i1], v2 = sc[i2];
    int r0 = 0, r1 = 0, r2 = 0;
    for (int j = 0; j < N_TOT; ++j) {
        const float s = sc[j];                       // LDS broadcast read
        r0 += (s > v0) || ((s == v0) && (j < i0));
        r1 += (s > v1) || ((s == v1) && (j < i1));
        r2 += (s > v2) || ((s == v2) && (j < i2));
    }
    if (r0 < N_MU) { perm[base + r0] = i0; wsc[base + r0] = v0; }
    if (r1 < N_MU) { perm[base + r1] = i1; wsc[base + r1] = v1; }
    if (r2 < N_MU) { perm[base + r2] = i2; wsc[base + r2] = v2; }
}

// ---------------------------------------------------------------------------
// Kernel 2: bandwidth kernel. One wave32 per output row (2 KB of mu + score).
// CDNA5 async-tensor path: per-lane GLOBAL_LOAD_ASYNC_TO_LDS_B128 stages the
// selected source row into LDS (512 B per instruction, no VGPR staging),
// S_WAIT_ASYNCCNT 0 synchronizes, then lane-strided b32 stores stream the
// 513-float row to global (rows are only 4 B aligned -> b32, fully coalesced).
// ---------------------------------------------------------------------------
__global__ __launch_bounds__(256) void gather_kernel(
    const float* __restrict__ cls_mu_queue,   // [N_CLASS, N_MU, OUT_DIM]
    const float* __restrict__ inp_mu,         // [K_IN, OUT_DIM]
    const int*   __restrict__ perm,           // [N_CLASS * N_MU]
    const float* __restrict__ wsc,            // [N_CLASS * N_MU]
    float*       __restrict__ out)            // [N_CLASS, N_MU, OUT_ROW]
{
    __shared__ float stage[ROWS_PER_BLOCK][OUT_DIM];   // 16 KB / block

    const int wid  = threadIdx.x >> 5;
    const int lane = threadIdx.x & 31;
    const int row  = blockIdx.x * ROWS_PER_BLOCK + wid;  // [0, 204800)
    const int c    = row >> 10;

    const int idx = perm[row];
    const float* src = (idx < N_MU)
        ? (cls_mu_queue + ((size_t)c * N_MU + idx) * OUT_DIM)
        : (inp_mu + (size_t)(idx - N_MU) * OUT_DIM);

    // Low 32 bits of a generic pointer into LDS == LDS byte offset (aperture
    // truncation, ISA 10.2) -> usable as the async-load LDS destination.
    const unsigned lds_base = (unsigned)(size_t)(&stage[wid][0]);

    // 4 x 16B per lane = 2 KB per wave, asynchronously into LDS.
#pragma unroll
    for (int k = 0; k < 4; ++k) {
        const float* g = src + (k * 128 + lane * 4);
        const unsigned l = lds_base + (unsigned)((k * 128 + lane * 4) * 4);
        asm volatile("global_load_async_to_lds_b128 %0, %1, off"
                     :: "v"(l), "v"(g) : "memory");
    }
    asm volatile("s_wait_asynccnt 0x0" ::: "memory");

    // LDS -> global, lane-strided (bank-conflict-free, coalesced stores).
    float* dst = out + (size_t)row * OUT_ROW;
#pragma unroll
    for (int s = 0; s < 16; ++s) {
        const int j = lane + 32 * s;
        dst[j] = stage[wid][j];
    }
    if (lane == 0) dst[OUT_DIM] = wsc[row];   // pack score column
}

// ---------------------------------------------------------------------------
extern "C" void kernel_launch(void* const* d_in, const int* in_sizes, int n_in,
                              void* d_out, int out_size, void* d_ws, size_t ws_size,
                              hipStream_t stream) {
    const float* cls_mu_queue = (const float*)d_in[0];
    const float* cls_sc_queue = (const float*)d_in[1];
    const float* inp_mu       = (const float*)d_in[2];
    const float* inp_sc       = (const float*)d_in[3];
    const int*   cls_idx      = (const int*)d_in[4];
    const int    n_idx        = in_sizes[4];

    float* out = (float*)d_out;

    // Workspace layout: perm [200*1024] i32, scores [200*1024] f32, flags [200] i32.
    int*   perm = (int*)d_ws;
    float* wsc  = (float*)(perm + N_CLASS * N_MU);
    int*   flag = (int*)(wsc + N_CLASS * N_MU);

    flags_kernel <<<1, 256, 0, stream>>>(cls_idx, n_idx, flag);
    select_kernel<<<N_CLASS, 512, 0, stream>>>(cls_sc_queue, inp_sc, flag, perm, wsc);
    gather_kernel<<<(N_CLASS * N_MU) / ROWS_PER_BLOCK, 256, 0, stream>>>(
        cls_mu_queue, inp_mu, perm, wsc, out);
}